// MemoryModule_58093727646163
// MI455X (gfx1250) — compile-verified
//
#include <hip/hip_runtime.h>
#include <math.h>

// ---------------- problem constants ----------------
#define HD   1024   // H
#define MD   4096   // M (memory rows)
#define WD   4096   // WORD
#define RN   4      // R read heads
#define NWH  2      // NW write heads
#define EPSF 0.001f

typedef float v2f __attribute__((ext_vector_type(2)));
typedef float v8f __attribute__((ext_vector_type(8)));

// ---------------- scratch layout (float offsets in d_ws) ----------------
#define WV_O     0          // write_vector    (2*4096)
#define EV_O     8192       // erase_vector    (2*4096)
#define WK_O     16384      // write_keys      (2*4096)
#define RK_O     24576      // read_keys       (4*4096)
#define SM_O     40960      // small values (64): fg[0..3] ag[4..5] wg[6..7] rm[8..27] ws[28..29] rs[30..33] kn[34..35] rkn[36..39]
#define DOT_O    41024      // write-key dot   (2*4096)
#define MNRM_O   49216      // mem norms       (4096)
#define USG_O    53312      // running usage   (4096)
#define RANK_O   57408      // rank (int view) (4096)
#define SRT_O    61504      // sorted u        (4096)
#define EXC_O    65600      // exclusive prod  (4096)
#define RDOT_O   69696      // read-key dot    (4*4096)
#define NNRM_O   86080      // new mem norms   (4096)
#define CW_O     90176      // content weights (4*4096)
#define FWD_O    106560     // fwd  [(r*2+w)*M + i]  (32768)
#define BWD_O    139328     // bwd  [(r*2+w)*M + j]  (32768)
#define BPART_O  172096     // bwd partials [((ib*2+w)*4+r)*M + j], ib<64 (2,097,152)

// ---------------- output layout (float offsets in d_out) ----------------
#define O_RWORDS 0
#define O_NMEM   16384
#define O_RW     16793600
#define O_WW     16809984
#define O_LINK   16818176
#define O_PREC   50372608
#define O_USAGE  50380800

__device__ __forceinline__ float sigf(float x) { return 1.0f / (1.0f + expf(-x)); }
__device__ __forceinline__ float wredsum(float v) {
  #pragma unroll
  for (int o = 16; o; o >>= 1) v += __shfl_xor(v, o, 32);
  return v;
}
__device__ __forceinline__ float dot4(float4 a, float4 b) {
  return a.x*b.x + a.y*b.y + a.z*b.z + a.w*b.w;
}

// ============ K0: small projections (34 rows) ============
__global__ void k_small_proj(const float* __restrict__ x,
    const float* Wfg, const float* bfg, const float* Wag, const float* bag,
    const float* Wwg, const float* bwg, const float* Wrm, const float* brm,
    const float* Wws, const float* bws, const float* Wrs, const float* brs,
    float* sm) {
  int wave = threadIdx.x >> 5, lane = threadIdx.x & 31;
  for (int row = wave; row < 34; row += 8) {
    const float* W; const float* b; int lr; int act;
    if      (row <  4) { W = Wfg; b = bfg; lr = row;      act = 1; }
    else if (row <  6) { W = Wag; b = bag; lr = row - 4;  act = 1; }
    else if (row <  8) { W = Wwg; b = bwg; lr = row - 6;  act = 1; }
    else if (row < 28) { W = Wrm; b = brm; lr = row - 8;  act = 1; }
    else if (row < 30) { W = Wws; b = bws; lr = row - 28; act = 0; }
    else               { W = Wrs; b = brs; lr = row - 30; act = 0; }
    const float* wr = W + (size_t)lr * HD;
    float s = 0.f;
    for (int k = lane; k < HD; k += 32) s += wr[k] * x[k];
    s = wredsum(s);
    if (lane == 0) {
      float v = s + b[lr];
      if (act) v = sigf(v);
      sm[row] = v;
    }
  }
}

// ============ K1: big projections via V_WMMA_F32_16X16X4_F32 ============
// out = W @ x (+bias, +act).  Per wave: 16 rows.  A = broadcast x-chunk (all 16
// M-rows equal), B = 16-col x 4-k tile of W (B[k][n] = W[row0+n][k0+k]).
// ISA layouts: A 16x4 f32: lane<16 -> K={0,1}; lane>=16 -> K={2,3} (VGPR sel low bit).
// D 16x16: VGPR0 lanes 0-15 = (M=0, N=lane) -> row j = row0 + lane.
__global__ void k_big_proj(const float* __restrict__ x,
    const float* Wwv, const float* bwv, const float* Wev, const float* bev,
    const float* Wwk, const float* bwk, const float* Wrk, const float* brk,
    float* wv, float* ev, float* wk, float* rk) {
  __shared__ float xs[HD];
  for (int i = threadIdx.x; i < HD; i += blockDim.x) xs[i] = x[i];
  __syncthreads();
  int wave = threadIdx.x >> 5, lane = threadIdx.x & 31;
  int tile = blockIdx.x * 8 + wave;     // 2560 tiles of 16 rows
  int row0 = tile * 16;
  const float* W; const float* b; float* dst; int act; int r0;
  if      (row0 <  8192) { W = Wwv; b = bwv; dst = wv; act = 0; r0 = row0;         }
  else if (row0 < 16384) { W = Wev; b = bev; dst = ev; act = 1; r0 = row0 - 8192;  }
  else if (row0 < 24576) { W = Wwk; b = bwk; dst = wk; act = 1; r0 = row0 - 16384; }
  else                   { W = Wrk; b = brk; dst = rk; act = 0; r0 = row0 - 24576; }
  int half = lane >> 4, lm = lane & 15;
  const float* wrow = W + (size_t)(r0 + lm) * HD;
  v8f c;
  #pragma unroll
  for (int q = 0; q < 8; q++) c[q] = 0.0f;
  for (int k0 = 0; k0 < HD; k0 += 4) {
    v2f a, bb;
    int kk = k0 + 2 * half;
    a[0]  = xs[kk];      a[1]  = xs[kk + 1];
    bb[0] = wrow[kk];    bb[1] = wrow[kk + 1];
    c = __builtin_amdgcn_wmma_f32_16x16x4_f32(false, a, false, bb, (short)0, c,
                                              false, false);
  }
  if (lane < 16) {
    float v = c[0] + b[r0 + lane];
    if (act) v = sigf(v);
    dst[r0 + lane] = v;
  }
}

// ============ K2: key norms (write_keys 2, read_keys 4) ============
__global__ void k_norms(const float* wk, const float* rk, float* sm) {
  int wave = threadIdx.x >> 5, lane = threadIdx.x & 31;
  if (wave >= 6) return;
  const float* p = (wave < 2) ? (wk + (size_t)wave * WD)
                              : (rk + (size_t)(wave - 2) * WD);
  float s = 0.f;
  for (int k = lane; k < WD; k += 32) { float v = p[k]; s += v * v; }
  s = wredsum(s);
  if (lane == 0) sm[34 + wave] = sqrtf(s);
}

// ============ K3: write-key dot with memory + memory norms ============
__global__ void k_memstats(const float* __restrict__ mem, const float* __restrict__ wk,
                           float* dotv, float* mnrm) {
  int wave = threadIdx.x >> 5, lane = threadIdx.x & 31;
  int m = blockIdx.x * 8 + wave;
  const float4* mr = (const float4*)(mem + (size_t)m * WD);
  const float4* k0 = (const float4*)wk;
  const float4* k1 = (const float4*)(wk + WD);
  float d0 = 0.f, d1 = 0.f, nn = 0.f;
  for (int q = lane; q < WD / 4; q += 32) {
    float4 mm = mr[q];
    d0 += dot4(mm, k0[q]);
    d1 += dot4(mm, k1[q]);
    nn += dot4(mm, mm);
  }
  d0 = wredsum(d0); d1 = wredsum(d1); nn = wredsum(nn);
  if (lane == 0) { dotv[m] = d0; dotv[MD + m] = d1; mnrm[m] = sqrtf(nn); }
}

// ============ K4: usage before head loop ============
__global__ void k_usage0(const float* pww, const float* prw, const float* pusage,
                         const float* sm, float* usg) {
  int m = blockIdx.x * blockDim.x + threadIdx.x;
  float wwt = 1.0f - pww[m] * pww[MD + m];
  float phi = 1.0f - (sm[0] * prw[m]) * (sm[1] * prw[MD + m]) *
                     (sm[2] * prw[2 * MD + m]) * (sm[3] * prw[3 * MD + m]);
  float pu = pusage[m];
  usg[m] = (pu + (1.0f - pu) * wwt) * phi;
}

// ============ K5a-d: allocation machinery (per head) ============
__global__ void k_zero_rank(int* rank) {
  rank[blockIdx.x * blockDim.x + threadIdx.x] = 0;
}

__global__ void k_rank_count(const float* usg, int* rank) {
  __shared__ float uj[256];
  int t = threadIdx.x;
  int jbase = blockIdx.y * 256;
  uj[t] = EPSF + (1.0f - EPSF) * usg[jbase + t];
  __syncthreads();
  int m = blockIdx.x * 256 + t;
  float um = EPSF + (1.0f - EPSF) * usg[m];
  int cnt = 0;
  #pragma unroll 8
  for (int jj = 0; jj < 256; jj++) {
    float v = uj[jj];
    int j = jbase + jj;
    cnt += (v < um) || (v == um && j < m);  // stable ascending argsort order
  }
  atomicAdd(&rank[m], cnt);  // integer atomic: exact & deterministic
}

__global__ void k_scatter(const float* usg, const int* rank, float* srt) {
  int m = blockIdx.x * blockDim.x + threadIdx.x;
  srt[rank[m]] = EPSF + (1.0f - EPSF) * usg[m];
}

__global__ void k_scan(const float* srt, float* exc) {  // 1 block x 1024
  __shared__ float s[MD];
  int t = threadIdx.x;
  #pragma unroll
  for (int c = 0; c < 4; c++) s[t + c * 1024] = srt[t + c * 1024];
  __syncthreads();
  for (int off = 1; off < MD; off <<= 1) {
    float tmp[4];
    #pragma unroll
    for (int c = 0; c < 4; c++) {
      int k = t + c * 1024;
      tmp[c] = (k >= off) ? s[k - off] * s[k] : s[k];
    }
    __syncthreads();
    #pragma unroll
    for (int c = 0; c < 4; c++) s[t + c * 1024] = tmp[c];
    __syncthreads();
  }
  #pragma unroll
  for (int c = 0; c < 4; c++) {
    int k = t + c * 1024;
    exc[k] = (k == 0) ? 1.0f : s[k - 1];
  }
}

// ============ K5e: per-head content softmax + allocation + usage + ww row ============
__global__ void k_head_update(int head, const float* sm, const float* dotv,
                              const float* mnrm, float* usg, const int* rank,
                              const float* exc, float* out_ww, float* out_usage) {
  __shared__ float red[1024];
  int t = threadIdx.x;
  float wsv = sm[28 + head];
  float beta = 1.0f + ((wsv > 20.0f) ? wsv : log1pf(expf(wsv)));
  float ag = sm[4 + head], wg = sm[6 + head], kn = sm[34 + head];
  const float* dh = dotv + (size_t)head * MD;
  // max
  float lm = -3.4e38f;
  #pragma unroll
  for (int c = 0; c < 4; c++) {
    int m = t + c * 1024;
    float s = dh[m] / (kn * mnrm[m] + EPSF) * beta;
    lm = fmaxf(lm, s);
  }
  red[t] = lm; __syncthreads();
  for (int o = 512; o; o >>= 1) { if (t < o) red[t] = fmaxf(red[t], red[t + o]); __syncthreads(); }
  float mx = red[0]; __syncthreads();
  // sum
  float ls = 0.f;
  #pragma unroll
  for (int c = 0; c < 4; c++) {
    int m = t + c * 1024;
    float s = dh[m] / (kn * mnrm[m] + EPSF) * beta;
    ls += expf(s - mx);
  }
  red[t] = ls; __syncthreads();
  for (int o = 512; o; o >>= 1) { if (t < o) red[t] += red[t + o]; __syncthreads(); }
  float sum = red[0]; __syncthreads();
  // update
  #pragma unroll
  for (int c = 0; c < 4; c++) {
    int m = t + c * 1024;
    float s  = dh[m] / (kn * mnrm[m] + EPSF) * beta;
    float cw = expf(s - mx) / sum;
    float u  = usg[m];
    float uu = EPSF + (1.0f - EPSF) * u;
    float a  = (1.0f - uu) * exc[rank[m]];
    out_ww[(size_t)head * MD + m] = wg * (ag * a + (1.0f - ag) * cw);
    float nu = u + (1.0f - u) * (ag * wg) * a;
    usg[m] = nu;
    if (head == NWH - 1) out_usage[m] = nu;
  }
}

// ============ K6: new_memory (fused erase/write + rdot + new norms) ============
__global__ void k_newmem(const float* __restrict__ mem, const float* __restrict__ ww,
                         const float* __restrict__ ev, const float* __restrict__ wv,
                         const float* __restrict__ rk, float* nmout, float* rdot,
                         float* nnrm) {
  int wave = threadIdx.x >> 5, lane = threadIdx.x & 31;
  int m = blockIdx.x * 8 + wave;
  float w0 = ww[m], w1 = ww[MD + m];
  const float4* mr = (const float4*)(mem + (size_t)m * WD);
  float4* dr = (float4*)(nmout + (size_t)m * WD);
  const float4* e0 = (const float4*)ev;
  const float4* e1 = (const float4*)(ev + WD);
  const float4* v0 = (const float4*)wv;
  const float4* v1 = (const float4*)(wv + WD);
  const float4* r0 = (const float4*)rk;
  const float4* r1 = (const float4*)(rk + WD);
  const float4* r2 = (const float4*)(rk + 2 * WD);
  const float4* r3 = (const float4*)(rk + 3 * WD);
  float a0 = 0.f, a1 = 0.f, a2 = 0.f, a3 = 0.f, nn = 0.f;
  for (int q = lane; q < WD / 4; q += 32) {
    float4 mm = mr[q], E0 = e0[q], E1 = e1[q], V0 = v0[q], V1 = v1[q];
    float4 nm;
    nm.x = mm.x * (1.0f - w0 * E0.x) * (1.0f - w1 * E1.x) + w0 * V0.x + w1 * V1.x;
    nm.y = mm.y * (1.0f - w0 * E0.y) * (1.0f - w1 * E1.y) + w0 * V0.y + w1 * V1.y;
    nm.z = mm.z * (1.0f - w0 * E0.z) * (1.0f - w1 * E1.z) + w0 * V0.z + w1 * V1.z;
    nm.w = mm.w * (1.0f - w0 * E0.w) * (1.0f - w1 * E1.w) + w0 * V0.w + w1 * V1.w;
    dr[q] = nm;
    a0 += dot4(r0[q], nm); a1 += dot4(r1[q], nm);
    a2 += dot4(r2[q], nm); a3 += dot4(r3[q], nm);
    nn += dot4(nm, nm);
  }
  a0 = wredsum(a0); a1 = wredsum(a1); a2 = wredsum(a2); a3 = wredsum(a3);
  nn = wredsum(nn);
  if (lane == 0) {
    rdot[m] = a0; rdot[MD + m] = a1; rdot[2 * MD + m] = a2; rdot[3 * MD + m] = a3;
    nnrm[m] = sqrtf(nn);
  }
}

// ============ K7: content read weights (softmax per r) ============
__global__ void k_content(const float* sm, const float* rdot, const float* nnrm,
                          float* cw) {
  __shared__ float red[1024];
  int r = blockIdx.x, t = threadIdx.x;
  float rkn = sm[36 + r], rs = sm[30 + r];
  const float* dh = rdot + (size_t)r * MD;
  float lm = -3.4e38f;
  #pragma unroll
  for (int c = 0; c < 4; c++) {
    int m = t + c * 1024;
    lm = fmaxf(lm, dh[m] / (rkn * nnrm[m]) * rs);
  }
  red[t] = lm; __syncthreads();
  for (int o = 512; o; o >>= 1) { if (t < o) red[t] = fmaxf(red[t], red[t + o]); __syncthreads(); }
  float mx = red[0]; __syncthreads();
  float ls = 0.f;
  #pragma unroll
  for (int c = 0; c < 4; c++) {
    int m = t + c * 1024;
    ls += expf(dh[m] / (rkn * nnrm[m]) * rs - mx);
  }
  red[t] = ls; __syncthreads();
  for (int o = 512; o; o >>= 1) { if (t < o) red[t] += red[t + o]; __syncthreads(); }
  float sum = red[0]; __syncthreads();
  #pragma unroll
  for (int c = 0; c < 4; c++) {
    int m = t + c * 1024;
    cw[(size_t)r * MD + m] = expf(dh[m] / (rkn * nnrm[m]) * rs - mx) / sum;
  }
}

// ============ K8: link update fused with bwd partial sums ============
#define ITILE 64
__global__ void k_link(const float* __restrict__ plink, const float* __restrict__ ww,
                       const float* __restrict__ pprec, const float* __restrict__ prw,
                       float* lnk, float* bpart) {
  int ib = blockIdx.x, w = blockIdx.y;
  int i0 = ib * ITILE;
  __shared__ float wwt[ITILE];
  __shared__ float prwt[RN][ITILE];
  int t = threadIdx.x;
  if (t < ITILE) wwt[t] = ww[(size_t)w * MD + i0 + t];
  { int r = t >> 6, il = t & 63; prwt[r][il] = prw[(size_t)r * MD + i0 + il]; }
  __syncthreads();
  for (int jt = t; jt < MD; jt += 256) {
    float wwj = ww[(size_t)w * MD + jt];
    float pj  = pprec[(size_t)w * MD + jt];
    float b0 = 0.f, b1 = 0.f, b2 = 0.f, b3 = 0.f;
    size_t base = ((size_t)w * MD + i0) * MD + jt;
    #pragma unroll 4
    for (int il = 0; il < ITILE; il++) {
      if ((il & 15) == 0 && il + 16 < ITILE)
        __builtin_prefetch(&plink[base + (size_t)(il + 16) * MD], 0, 1);
      int i = i0 + il;
      float L = plink[base + (size_t)il * MD];
      float val = (i == jt) ? 0.0f : ((1.0f - wwt[il] - wwj) * L + wwt[il] * pj);
      lnk[base + (size_t)il * MD] = val;
      b0 += prwt[0][il] * val; b1 += prwt[1][il] * val;
      b2 += prwt[2][il] * val; b3 += prwt[3][il] * val;
    }
    size_t pb = (((size_t)ib * NWH + w) * RN) * MD + jt;
    bpart[pb] = b0; bpart[pb + MD] = b1; bpart[pb + 2 * MD] = b2; bpart[pb + 3 * MD] = b3;
  }
}

// ============ K9: precedence ============
__global__ void k_prec(const float* ww, const float* pprec, float* prec) {
  __shared__ float red[1024];
  __shared__ float S[NWH];
  int t = threadIdx.x;
  for (int w = 0; w < NWH; w++) {
    float s = 0.f;
    #pragma unroll
    for (int c = 0; c < 4; c++) s += ww[(size_t)w * MD + t + c * 1024];
    red[t] = s; __syncthreads();
    for (int o = 512; o; o >>= 1) { if (t < o) red[t] += red[t + o]; __syncthreads(); }
    if (t == 0) S[w] = red[0];
    __syncthreads();
  }
  for (int w = 0; w < NWH; w++)
    #pragma unroll
    for (int c = 0; c < 4; c++) {
      int m = t + c * 1024;
      prec[(size_t)w * MD + m] =
          (1.0f - S[w]) * pprec[(size_t)w * MD + m] + ww[(size_t)w * MD + m];
    }
}

// ============ K10: fwd weights from new link ============
__global__ void k_fwd(const float* __restrict__ lnk, const float* __restrict__ prw,
                      float* fwd) {
  int wave = threadIdx.x >> 5, lane = threadIdx.x & 31;
  int gid = blockIdx.x * 8 + wave;   // 0..8191
  int w = gid >> 12, i = gid & (MD - 1);
  const float4* lr = (const float4*)(lnk + ((size_t)w * MD + i) * MD);
  const float4* p0 = (const float4*)prw;
  const float4* p1 = (const float4*)(prw + MD);
  const float4* p2 = (const float4*)(prw + 2 * MD);
  const float4* p3 = (const float4*)(prw + 3 * MD);
  float f0 = 0.f, f1 = 0.f, f2 = 0.f, f3 = 0.f;
  for (int q = lane; q < MD / 4; q += 32) {
    float4 L = lr[q];
    f0 += dot4(p0[q], L); f1 += dot4(p1[q], L);
    f2 += dot4(p2[q], L); f3 += dot4(p3[q], L);
  }
  f0 = wredsum(f0); f1 = wredsum(f1); f2 = wredsum(f2); f3 = wredsum(f3);
  if (lane == 0) {
    fwd[(size_t)(0 * NWH + w) * MD + i] = f0;
    fwd[(size_t)(1 * NWH + w) * MD + i] = f1;
    fwd[(size_t)(2 * NWH + w) * MD + i] = f2;
    fwd[(size_t)(3 * NWH + w) * MD + i] = f3;
  }
}

// ============ K11: reduce bwd partials (fixed order, deterministic) ============
__global__ void k_bwd_reduce(const float* bpart, float* bwd) {
  int t = blockIdx.x * blockDim.x + threadIdx.x;  // 0..32767
  int j = t & (MD - 1);
  int rw = t >> 12;            // r*2 + w
  int r = rw >> 1, w = rw & 1;
  float s = 0.f;
  for (int ib = 0; ib < 64; ib++)
    s += bpart[(((size_t)ib * NWH + w) * RN + r) * MD + j];
  bwd[t] = s;
}

// ============ K12: combine read weights ============
__global__ void k_readw(const float* sm, const float* fwd, const float* bwd,
                        const float* cw, float* out_rw) {
  int t = blockIdx.x * blockDim.x + threadIdx.x;  // 0..16383
  int r = t >> 12, m = t & (MD - 1);
  const float* rm = sm + 8 + r * 5;
  float v = rm[0] * bwd[(size_t)(r * NWH + 0) * MD + m] +
            rm[1] * bwd[(size_t)(r * NWH + 1) * MD + m] +
            rm[2] * fwd[(size_t)(r * NWH + 0) * MD + m] +
            rm[3] * fwd[(size_t)(r * NWH + 1) * MD + m] +
            rm[4] * cw[(size_t)r * MD + m];
  out_rw[(size_t)r * MD + m] = v;
}

// ============ K13: read_words ============
__global__ void k_readwords(const float* __restrict__ rw, const float* __restrict__ nm,
                            float* out_words) {
  __shared__ float red[8][32][RN];
  int dl = threadIdx.x & 31, ms = threadIdx.x >> 5;
  int d = blockIdx.x * 32 + dl;
  float acc[RN] = {0.f, 0.f, 0.f, 0.f};
  for (int m = ms; m < MD; m += 8) {
    float v = nm[(size_t)m * WD + d];
    acc[0] += rw[m] * v;
    acc[1] += rw[MD + m] * v;
    acc[2] += rw[2 * MD + m] * v;
    acc[3] += rw[3 * MD + m] * v;
  }
  #pragma unroll
  for (int r = 0; r < RN; r++) red[ms][dl][r] = acc[r];
  __syncthreads();
  if (ms == 0) {
    #pragma unroll
    for (int r = 0; r < RN; r++) {
      float s = red[0][dl][r];
      #pragma unroll
      for (int k = 1; k < 8; k++) s += red[k][dl][r];
      out_words[(size_t)r * WD + d] = s;
    }
  }
}

// ================= launcher =================
extern "C" void kernel_launch(void* const* d_in, const int* in_sizes, int n_in,
                              void* d_out, int out_size, void* d_ws, size_t ws_size,
                              hipStream_t stream) {
  (void)in_sizes; (void)n_in; (void)out_size; (void)ws_size;
  const float* x      = (const float*)d_in[0];
  const float* memory = (const float*)d_in[1];
  const float* prw    = (const float*)d_in[2];
  const float* pww    = (const float*)d_in[3];
  const float* plink  = (const float*)d_in[4];
  const float* pprec  = (const float*)d_in[5];
  const float* pusage = (const float*)d_in[6];
  const float* Wwv = (const float*)d_in[7];  const float* bwv = (const float*)d_in[8];
  const float* Wev = (const float*)d_in[9];  const float* bev = (const float*)d_in[10];
  const float* Wfg = (const float*)d_in[11]; const float* bfg = (const float*)d_in[12];
  const float* Wag = (const float*)d_in[13]; const float* bag = (const float*)d_in[14];
  const float* Wwg = (const float*)d_in[15]; const float* bwg = (const float*)d_in[16];
  const float* Wrm = (const float*)d_in[17]; const float* brm = (const float*)d_in[18];
  const float* Wwk = (const float*)d_in[19]; const float* bwk = (const float*)d_in[20];
  const float* Wws = (const float*)d_in[21]; const float* bws = (const float*)d_in[22];
  const float* Wrk = (const float*)d_in[23]; const float* brk = (const float*)d_in[24];
  const float* Wrs = (const float*)d_in[25]; const float* brs = (const float*)d_in[26];

  float* out = (float*)d_out;
  float* ws  = (float*)d_ws;
  float* sm  = ws + SM_O;
  int*   rank = (int*)(ws + RANK_O);

  k_small_proj<<<1, 256, 0, stream>>>(x, Wfg, bfg, Wag, bag, Wwg, bwg, Wrm, brm,
                                      Wws, bws, Wrs, brs, sm);
  k_big_proj<<<320, 256, 0, stream>>>(x, Wwv, bwv, Wev, bev, Wwk, bwk, Wrk, brk,
                                      ws + WV_O, ws + EV_O, ws + WK_O, ws + RK_O);
  k_norms<<<1, 256, 0, stream>>>(ws + WK_O, ws + RK_O, sm);
  k_memstats<<<512, 256, 0, stream>>>(memory, ws + WK_O, ws + DOT_O, ws + MNRM_O);
  k_usage0<<<16, 256, 0, stream>>>(pww, prw, pusage, sm, ws + USG_O);

  for (int head = 0; head < NWH; head++) {
    k_zero_rank<<<16, 256, 0, stream>>>(rank);
    k_rank_count<<<dim3(16, 16), 256, 0, stream>>>(ws + USG_O, rank);
    k_scatter<<<16, 256, 0, stream>>>(ws + USG_O, rank, ws + SRT_O);
    k_scan<<<1, 1024, 0, stream>>>(ws + SRT_O, ws + EXC_O);
    k_head_update<<<1, 1024, 0, stream>>>(head, sm, ws + DOT_O, ws + MNRM_O,
                                          ws + USG_O, rank, ws + EXC_O,
                                          out + O_WW, out + O_USAGE);
  }

  k_newmem<<<512, 256, 0, stream>>>(memory, out + O_WW, ws + EV_O, ws + WV_O,
                                    ws + RK_O, out + O_NMEM, ws + RDOT_O,
                                    ws + NNRM_O);
  k_content<<<RN, 1024, 0, stream>>>(sm, ws + RDOT_O, ws + NNRM_O, ws + CW_O);
  k_link<<<dim3(64, NWH), 256, 0, stream>>>(plink, out + O_WW, pprec, prw,
                                            out + O_LINK, ws + BPART_O);
  k_prec<<<1, 1024, 0, stream>>>(out + O_WW, pprec, out + O_PREC);
  k_fwd<<<1024, 256, 0, stream>>>(out + O_LINK, prw, ws + FWD_O);
  k_bwd_reduce<<<128, 256, 0, stream>>>(ws + BPART_O, ws + BWD_O);
  k_readw<<<64, 256, 0, stream>>>(sm, ws + FWD_O, ws + BWD_O, ws + CW_O,
                                  out + O_RW);
  k_readwords<<<128, 256, 0, stream>>>(out + O_RW, out + O_NMEM, out + O_RWORDS);
}